// SimplicialConvolution2_57432302682843
// MI455X (gfx1250) — compile-verified
//
#include <hip/hip_runtime.h>

#define MS   100000          // simplices
#define NNZC 800000          // nonzeros per Laplacian
#define NB   2               // batch
#define CI   64              // C_in
#define CO   64              // C_out
#define NSRC 5               // [x, Ll x, Ll^2 x, Lu x, Lu^2 x]
#define BCR  (NB * CI)       // 128 combined (batch, channel) rows
#define KW   (NSRC * CI)     // 320 effective K
#define MT16 (MS / 16)       // 6250 m-tiles (MS divisible by 16)
#define LDSP 324             // LDS pitch for W: 324 % 64 banks = 4 -> conflict-free

typedef __attribute__((ext_vector_type(2))) float v2f;
typedef __attribute__((ext_vector_type(8))) float v8f;

// ---------------------------------------------------------------- zero scratch
__global__ void zero_ws_kernel(float4* __restrict__ p, int n4) {
  int i = blockIdx.x * blockDim.x + threadIdx.x;
  if (i < n4) {
    float4 z; z.x = 0.f; z.y = 0.f; z.z = 0.f; z.w = 0.f;
    p[i] = z;
  }
}

// -------------------------------------------------- build W (CO x KW) from theta
// W[o][s*CI + i] = theta[o][i][ksel[s]]  (+ theta[o][i][3] for s==0, since the
// k=0 taps of both Laplacian stacks are both 'x')
__global__ void build_w_kernel(const float* __restrict__ theta, float* __restrict__ W) {
  int idx = blockIdx.x * blockDim.x + threadIdx.x;
  if (idx >= CO * KW) return;
  int o = idx / KW;
  int k = idx - o * KW;
  int s = k / CI;
  int i = k - s * CI;
  const int ksel[NSRC] = {0, 1, 2, 4, 5};
  float w = theta[(o * CI + i) * 6 + ksel[s]];
  if (s == 0) w += theta[(o * CI + i) * 6 + 3];
  W[idx] = w;
}

// ---------------------------------------------------------------- SpMM (COO)
// dst[bc][row] += val * src[bc][col] over all 128 (batch,channel) planes.
// One thread per nonzero: index loads are coalesced; gathers/atomics stay
// inside one 400KB plane row per iteration (L2-resident on 192MB L2).
__global__ void spmm_kernel(const int* __restrict__ rows, const int* __restrict__ cols,
                            const float* __restrict__ vals,
                            const float* __restrict__ src, float* __restrict__ dst) {
  int e = blockIdx.x * blockDim.x + threadIdx.x;
  if (e >= NNZC) return;
  const int   r = rows[e];
  const int   c = cols[e];
  const float v = vals[e];
  const float* sp = src + c;
  float*       dp = dst + r;
#pragma unroll 8
  for (int bc = 0; bc < BCR; ++bc) {
    unsafeAtomicAdd(dp + (size_t)bc * MS, v * sp[(size_t)bc * MS]);
  }
}

// ------------------------------------------------------------- WMMA GEMM
// Per wave: 64(out) x 16(m) tile of one batch.
// Y[o][m] = sum_k W[o][k] * S_k[m],  K = 320 split across 5 source planes.
__global__ __launch_bounds__(256)
void gemm_wmma_kernel(const float* __restrict__ x,
                      const float* __restrict__ P1, const float* __restrict__ P2,
                      const float* __restrict__ Q1, const float* __restrict__ Q2,
                      const float* __restrict__ W,  const float* __restrict__ bias,
                      float* __restrict__ y) {
  __shared__ float ldsW[CO * LDSP];
  for (int idx = threadIdx.x; idx < CO * KW; idx += blockDim.x) {
    int o = idx / KW, k = idx - o * KW;
    ldsW[o * LDSP + k] = W[idx];
  }
  __syncthreads();

  const int waveId = blockIdx.x * (blockDim.x >> 5) + (threadIdx.x >> 5);
  if (waveId >= NB * MT16) return;   // wave-uniform exit: EXEC all-ones for WMMA
  const int b    = waveId / MT16;
  const int mt   = waveId - b * MT16;
  const int m0   = mt * 16;
  const int lane = threadIdx.x & 31;
  const int hi   = lane >> 4;        // 0: K+{0,1}, 1: K+{2,3}
  const int n    = lane & 15;        // A: row M; B: col N; D: col N

  const size_t bstride = (size_t)CI * MS;
  const float* srcs[NSRC] = { x + b * bstride, P1 + b * bstride, P2 + b * bstride,
                              Q1 + b * bstride, Q2 + b * bstride };

  v8f acc0 = {}, acc1 = {}, acc2 = {}, acc3 = {};

#pragma unroll
  for (int s = 0; s < NSRC; ++s) {
    const float* S = srcs[s] + m0 + n;
    for (int ki = 0; ki < CI; ki += 4) {
      const int kb = ki + 2 * hi;
      // B-tile (4x16): VGPR0 = K=kb row, VGPR1 = K=kb+1 row, cols m-contiguous
      v2f bm;
      bm.x = S[(size_t)kb * MS];
      bm.y = S[(size_t)(kb + 1) * MS];
      // A-tiles (16x4) from LDS: lane row o = r*16+n, 2 consecutive K per lane
      const int   kk = s * CI + kb;
      const float* wl = &ldsW[n * LDSP + kk];
      v2f a0, a1, a2, a3;
      a0.x = wl[0];              a0.y = wl[1];
      a1.x = wl[16 * LDSP];      a1.y = wl[16 * LDSP + 1];
      a2.x = wl[32 * LDSP];      a2.y = wl[32 * LDSP + 1];
      a3.x = wl[48 * LDSP];      a3.y = wl[48 * LDSP + 1];
      acc0 = __builtin_amdgcn_wmma_f32_16x16x4_f32(false, a0, false, bm, (short)0, acc0, false, false);
      acc1 = __builtin_amdgcn_wmma_f32_16x16x4_f32(false, a1, false, bm, (short)0, acc1, false, false);
      acc2 = __builtin_amdgcn_wmma_f32_16x16x4_f32(false, a2, false, bm, (short)0, acc2, false, false);
      acc3 = __builtin_amdgcn_wmma_f32_16x16x4_f32(false, a3, false, bm, (short)0, acc3, false, false);
    }
  }

  // D layout: VGPR j -> row (j + 8*hi), col n. Rows are out-channels -> m-contiguous stores.
  float* yb = y + (size_t)b * CO * MS + m0 + n;
#pragma unroll
  for (int j = 0; j < 8; ++j) {
    { int o =  0 + j + 8 * hi; yb[(size_t)o * MS] = acc0[j] + bias[o]; }
    { int o = 16 + j + 8 * hi; yb[(size_t)o * MS] = acc1[j] + bias[o]; }
    { int o = 32 + j + 8 * hi; yb[(size_t)o * MS] = acc2[j] + bias[o]; }
    { int o = 48 + j + 8 * hi; yb[(size_t)o * MS] = acc3[j] + bias[o]; }
  }
}

// ---------------------------------------------------------------- launch
extern "C" void kernel_launch(void* const* d_in, const int* in_sizes, int n_in,
                              void* d_out, int out_size, void* d_ws, size_t ws_size,
                              hipStream_t stream) {
  const int*   Ll_rows = (const int*)d_in[0];
  const int*   Ll_cols = Ll_rows + NNZC;
  const float* Ll_vals = (const float*)d_in[1];
  const int*   Lu_rows = (const int*)d_in[2];
  const int*   Lu_cols = Lu_rows + NNZC;
  const float* Lu_vals = (const float*)d_in[3];
  const float* x       = (const float*)d_in[4];
  const float* theta   = (const float*)d_in[5];
  const float* bias    = (const float*)d_in[6];
  float*       y       = (float*)d_out;

  const size_t plane = (size_t)BCR * MS;        // 12.8M floats per power buffer
  float* P1 = (float*)d_ws;
  float* P2 = P1 + plane;
  float* Q1 = P2 + plane;
  float* Q2 = Q1 + plane;
  float* W  = Q2 + plane;                        // 64*320 floats

  // 1) zero the four power accumulators
  {
    int n4 = (int)(4 * plane / 4);
    zero_ws_kernel<<<(n4 + 255) / 256, 256, 0, stream>>>((float4*)P1, n4);
  }
  // 2) fold theta into W (64 x 320)
  build_w_kernel<<<(CO * KW + 255) / 256, 256, 0, stream>>>(theta, W);

  // 3) four SpMM passes (serially dependent pairs)
  const int sblocks = (NNZC + 255) / 256;
  spmm_kernel<<<sblocks, 256, 0, stream>>>(Ll_rows, Ll_cols, Ll_vals, x,  P1);
  spmm_kernel<<<sblocks, 256, 0, stream>>>(Ll_rows, Ll_cols, Ll_vals, P1, P2);
  spmm_kernel<<<sblocks, 256, 0, stream>>>(Lu_rows, Lu_cols, Lu_vals, x,  Q1);
  spmm_kernel<<<sblocks, 256, 0, stream>>>(Lu_rows, Lu_cols, Lu_vals, Q1, Q2);

  // 4) fused WMMA contraction + bias
  const int tiles   = NB * MT16;                 // 12500 waves
  const int gblocks = (tiles + 7) / 8;           // 8 waves / block
  gemm_wmma_kernel<<<gblocks, 256, 0, stream>>>(x, P1, P2, Q1, Q2, W, bias, y);
}